// Projection_7447473292035
// MI455X (gfx1250) — compile-verified
//
#include <hip/hip_runtime.h>
#include <hip/hip_bf16.h>
#include <math.h>

// ---------------------------------------------------------------------------
// Projection: A = einsum('icab,ni->ncab', antiherm(M), dX);  out = expm(A)
// via scaling-and-squaring with extra scaling + degree-10 Taylor (no LU solve).
// fp32 WMMA (V_WMMA_F32_16X16X4_F32) keeps full precision through ~20
// dependent matmuls + 2^s squaring amplification.
// ---------------------------------------------------------------------------

typedef float v2f __attribute__((ext_vector_type(2)));
typedef float v8f __attribute__((ext_vector_type(8)));

#define LDPAD 65      // 64 + 1 float padding: column reads hit distinct banks
#define KT 10         // Taylor order
#define EXTRA_S 4     // extra halvings -> ||B||_1 <= 1/16 before Taylor
#define MAX_S 30

static __device__ __forceinline__ v8f wmma4(v2f a, v2f b, v8f c) {
  // D = A(16x4, f32) x B(4x16, f32) + C(16x16, f32)
  return __builtin_amdgcn_wmma_f32_16x16x4_f32(
      /*neg_a=*/false, a, /*neg_b=*/false, b,
      /*c_mod=*/(short)0, c, /*reuse_a=*/false, /*reuse_b=*/false);
}

// --------------------------- Kernel 1: anti-Hermitian ----------------------
// Ah[i,c,a,b] = 0.5*((Re[a,b]-Re[b,a]) + j*(Im[a,b]+Im[b,a])), stored as two
// real planes with layout [i][c*4096 + a*64 + b] == same flat index as mm_*.
__global__ __launch_bounds__(256) void antiherm_kernel(
    const float* __restrict__ mm_real, const float* __restrict__ mm_imag,
    float* __restrict__ AhRe, float* __restrict__ AhIm)
{
  const int idx = blockIdx.x * 256 + threadIdx.x;   // 128*2*64*64 = 1,048,576
  const int ab = idx & 4095;
  const int a = ab >> 6, b = ab & 63;
  const int icbase = idx - ab;                      // (i*2+c)*4096
  const float re_ab = mm_real[icbase + a * 64 + b];
  const float re_ba = mm_real[icbase + b * 64 + a];
  const float im_ab = mm_imag[icbase + a * 64 + b];
  const float im_ba = mm_imag[icbase + b * 64 + a];
  AhRe[idx] = 0.5f * (re_ab - re_ba);
  AhIm[idx] = 0.5f * (im_ab + im_ba);
}

// --------------------------- Kernel 2: WMMA GEMM ---------------------------
// A(1024 x 8192 complex) = dX(1024x128, real) @ Ah(128 x 8192, complex).
// Real A-operand shared by both re/im accumulators. Output written as
// interleaved complex64 (numpy layout) straight into d_out.
__global__ __launch_bounds__(256) void proj_gemm_kernel(
    const float* __restrict__ dX,     // 1024 x 128
    const float* __restrict__ AhRe,   // 128 x 8192
    const float* __restrict__ AhIm,   // 128 x 8192
    float2* __restrict__ out)         // 1024 x 8192 (complex)
{
  const int tid  = threadIdx.x;
  const int lane = tid & 31;
  const int wave = tid >> 5;
  const int ti   = wave >> 1;          // 16x16 tile row within 64x64 WG tile
  const int tj0  = (wave & 1) * 2;     // two adjacent tile cols per wave
  const int r15  = lane & 15;
  const int kh   = (lane >> 4) << 1;   // K sub-offset per A/B f32 layout

  const int m0 = blockIdx.y * 64 + ti * 16;
  const int j0 = blockIdx.x * 64 + tj0 * 16 + r15;
  const int j1 = j0 + 16;

  v8f accRe0 = {0,0,0,0,0,0,0,0}, accIm0 = {0,0,0,0,0,0,0,0};
  v8f accRe1 = {0,0,0,0,0,0,0,0}, accIm1 = {0,0,0,0,0,0,0,0};

  const float* dxrow = dX + (size_t)(m0 + r15) * 128;

  for (int k0 = 0; k0 < 128; k0 += 4) {
    const int ka = k0 + kh;
    v2f a, br0, bi0, br1, bi1;
    a.x = dxrow[ka];  a.y = dxrow[ka + 1];
    const size_t o0 = (size_t)ka * 8192, o1 = o0 + 8192;
    br0.x = AhRe[o0 + j0]; br0.y = AhRe[o1 + j0];
    bi0.x = AhIm[o0 + j0]; bi0.y = AhIm[o1 + j0];
    br1.x = AhRe[o0 + j1]; br1.y = AhRe[o1 + j1];
    bi1.x = AhIm[o0 + j1]; bi1.y = AhIm[o1 + j1];
    accRe0 = wmma4(a, br0, accRe0);
    accIm0 = wmma4(a, bi0, accIm0);
    accRe1 = wmma4(a, br1, accRe1);
    accIm1 = wmma4(a, bi1, accIm1);
  }

  const int rbase = m0 + ((lane >> 4) << 3);   // C layout: VGPR v -> M = v | v+8
  #pragma unroll
  for (int v = 0; v < 8; ++v) {
    const size_t r = (size_t)(rbase + v);
    out[r * 8192 + j0] = make_float2(accRe0[v], accIm0[v]);
    out[r * 8192 + j1] = make_float2(accRe1[v], accIm1[v]);
  }
}

// --------------------- Kernel 3: per-matrix complex expm -------------------
struct CAcc { v8f re0, im0, re1, im1; };

// Complex 64x64 matmul from LDS planes into registers. 8 waves x 2 tiles.
// re = A_re*B_re - A_im*B_im (two accs, VALU subtract: f32 WMMA has no A-neg)
// im = A_re*B_im + A_im*B_re (chained accumulation)
static __device__ __forceinline__ CAcc cmm64(
    const float* __restrict__ Ar, const float* __restrict__ Ai,
    const float* __restrict__ Br, const float* __restrict__ Bi, int tid)
{
  const int lane = tid & 31;
  const int wave = tid >> 5;
  const int ti   = wave >> 1;
  const int tj0  = (wave & 1) * 2;
  const int r15  = lane & 15;
  const int kh   = (lane >> 4) << 1;

  v8f aRR0 = {0,0,0,0,0,0,0,0}, aII0 = {0,0,0,0,0,0,0,0}, aIm0 = {0,0,0,0,0,0,0,0};
  v8f aRR1 = {0,0,0,0,0,0,0,0}, aII1 = {0,0,0,0,0,0,0,0}, aIm1 = {0,0,0,0,0,0,0,0};

  const int arow = (ti * 16 + r15) * LDPAD;
  const int c0 = tj0 * 16 + r15;
  const int c1 = c0 + 16;

  for (int k0 = 0; k0 < 64; k0 += 4) {
    const int ka = k0 + kh;
    v2f ar, ai, br0, bi0, br1, bi1;
    ar.x = Ar[arow + ka]; ar.y = Ar[arow + ka + 1];
    ai.x = Ai[arow + ka]; ai.y = Ai[arow + ka + 1];
    const int b0 = ka * LDPAD, b1 = b0 + LDPAD;
    br0.x = Br[b0 + c0]; br0.y = Br[b1 + c0];
    bi0.x = Bi[b0 + c0]; bi0.y = Bi[b1 + c0];
    br1.x = Br[b0 + c1]; br1.y = Br[b1 + c1];
    bi1.x = Bi[b0 + c1]; bi1.y = Bi[b1 + c1];
    aRR0 = wmma4(ar, br0, aRR0);
    aII0 = wmma4(ai, bi0, aII0);
    aIm0 = wmma4(ar, bi0, aIm0);
    aIm0 = wmma4(ai, br0, aIm0);
    aRR1 = wmma4(ar, br1, aRR1);
    aII1 = wmma4(ai, bi1, aII1);
    aIm1 = wmma4(ar, bi1, aIm1);
    aIm1 = wmma4(ai, br1, aIm1);
  }
  CAcc o;
  o.re0 = aRR0 - aII0;  o.im0 = aIm0;
  o.re1 = aRR1 - aII1;  o.im1 = aIm1;
  return o;
}

static __device__ __forceinline__ void store_acc(
    float* __restrict__ Dr, float* __restrict__ Di,
    const CAcc& t, int tid, float mul, bool addI)
{
  const int lane = tid & 31;
  const int wave = tid >> 5;
  const int ti   = wave >> 1;
  const int tj0  = (wave & 1) * 2;
  const int r15  = lane & 15;
  const int c0 = tj0 * 16 + r15;
  const int c1 = c0 + 16;
  const int rbase = ti * 16 + ((lane >> 4) << 3);
  #pragma unroll
  for (int v = 0; v < 8; ++v) {
    const int r = rbase + v;
    const float d0 = (addI && r == c0) ? 1.0f : 0.0f;
    const float d1 = (addI && r == c1) ? 1.0f : 0.0f;
    Dr[r * LDPAD + c0] = t.re0[v] * mul + d0;
    Di[r * LDPAD + c0] = t.im0[v] * mul;
    Dr[r * LDPAD + c1] = t.re1[v] * mul + d1;
    Di[r * LDPAD + c1] = t.im1[v] * mul;
  }
}

__global__ __launch_bounds__(256) void expm_kernel(float2* __restrict__ out)
{
  __shared__ float Br_[64 * LDPAD], Bi_[64 * LDPAD];   // scaled input B
  __shared__ float Tr_[64 * LDPAD], Ti_[64 * LDPAD];   // running result T
  __shared__ float colsum[64];
  __shared__ float scale_sh;
  __shared__ int   stot_sh;

  const int tid = threadIdx.x;
  float2* base = out + (size_t)blockIdx.x * 4096;      // one (n,c) matrix / WG

  // Load A into T planes
  for (int idx = tid; idx < 4096; idx += 256) {
    const int a = idx >> 6, b = idx & 63;
    const float2 v = base[idx];
    Tr_[a * LDPAD + b] = v.x;
    Ti_[a * LDPAD + b] = v.y;
  }
  __syncthreads();

  // Induced 1-norm: max over columns of sum_a |A[a,b]|
  if (tid < 64) {
    float s = 0.f;
    for (int a = 0; a < 64; ++a) {
      const float re = Tr_[a * LDPAD + tid], im = Ti_[a * LDPAD + tid];
      s += sqrtf(re * re + im * im);
    }
    colsum[tid] = s;
  }
  __syncthreads();
  if (tid == 0) {
    float nrm = 0.f;
    for (int b = 0; b < 64; ++b) nrm = fmaxf(nrm, colsum[b]);
    int sref = 0;
    if (nrm > 1.0f) { sref = (int)ceilf(log2f(nrm)); if (sref < 0) sref = 0; }
    int stot = sref + EXTRA_S;
    if (stot > MAX_S) stot = MAX_S;
    stot_sh  = stot;
    scale_sh = exp2f(-(float)stot);
  }
  __syncthreads();
  const int   stot  = stot_sh;
  const float scale = scale_sh;

  // B = A * 2^-stot ; T = I + B/KT
  const float invKT = 1.0f / (float)KT;
  for (int idx = tid; idx < 4096; idx += 256) {
    const int a = idx >> 6, b = idx & 63;
    const float br = Tr_[a * LDPAD + b] * scale;
    const float bi = Ti_[a * LDPAD + b] * scale;
    Br_[a * LDPAD + b] = br;
    Bi_[a * LDPAD + b] = bi;
    Tr_[a * LDPAD + b] = br * invKT + ((a == b) ? 1.0f : 0.0f);
    Ti_[a * LDPAD + b] = bi * invKT;
  }
  __syncthreads();

  // Horner: T <- I + (B @ T)/k, k = KT-1 .. 1
  for (int k = KT - 1; k >= 1; --k) {
    const CAcc p = cmm64(Br_, Bi_, Tr_, Ti_, tid);
    __syncthreads();                       // all reads of T done
    store_acc(Tr_, Ti_, p, tid, 1.0f / (float)k, /*addI=*/true);
    __syncthreads();
  }

  // Squarings: T <- T @ T, stot times (uniform per workgroup -> EXEC all-1s)
  for (int q = 0; q < stot; ++q) {
    const CAcc p = cmm64(Tr_, Ti_, Tr_, Ti_, tid);
    __syncthreads();
    store_acc(Tr_, Ti_, p, tid, 1.0f, /*addI=*/false);
    __syncthreads();
  }

  // Write E back over A (interleaved complex64)
  for (int idx = tid; idx < 4096; idx += 256) {
    const int a = idx >> 6, b = idx & 63;
    float2 v;
    v.x = Tr_[a * LDPAD + b];
    v.y = Ti_[a * LDPAD + b];
    base[idx] = v;
  }
}

// ------------------------------- launcher ----------------------------------
extern "C" void kernel_launch(void* const* d_in, const int* in_sizes, int n_in,
                              void* d_out, int out_size, void* d_ws, size_t ws_size,
                              hipStream_t stream) {
  const float* dX      = (const float*)d_in[0];   // (1024, 128)
  const float* mm_real = (const float*)d_in[1];   // (128, 2, 64, 64)
  const float* mm_imag = (const float*)d_in[2];   // (128, 2, 64, 64)

  float* AhRe = (float*)d_ws;                     // 128 x 8192 (4 MB)
  float* AhIm = AhRe + (size_t)128 * 8192;        // 128 x 8192 (4 MB)
  float2* out = (float2*)d_out;                   // 1024 x 8192 complex

  antiherm_kernel<<<4096, 256, 0, stream>>>(mm_real, mm_imag, AhRe, AhIm);

  dim3 g2(128, 16);                               // N/64, M/64
  proj_gemm_kernel<<<g2, 256, 0, stream>>>(dX, AhRe, AhIm, out);

  expm_kernel<<<2048, 256, 0, stream>>>(out);     // one WG per (n,c) matrix
}